// TestRotConv_48576080117803
// MI455X (gfx1250) — compile-verified
//
#include <hip/hip_runtime.h>
#include <hip/hip_bf16.h>

// Problem constants (from reference)
#define N_NODES 100000
#define N_EDGES 1200000
#define CH      32        // C
#define CD      64        // C*D
#define NLAYERS 4
#define ECOMP   8
#define NGRAPH  256

typedef float v2f __attribute__((ext_vector_type(2)));
typedef float v8f __attribute__((ext_vector_type(8)));

#if defined(__gfx1250__) && __has_builtin(__builtin_amdgcn_global_load_async_to_lds_b128) && __has_builtin(__builtin_amdgcn_s_wait_asynccnt)
#define HAVE_ASYNC_LDS 1
#else
#define HAVE_ASYNC_LDS 0
#endif

#if HAVE_ASYNC_LDS
// The builtin takes pointers to a GCC-style int32x4 vector in AS1 (global)
// and AS3 (LDS), per the clang diagnostic from the probe round.
typedef int async_v4i __attribute__((vector_size(4 * sizeof(int))));
#define GLOBAL_AS(p) ((__attribute__((address_space(1))) async_v4i*)(p))
#define LDS_AS(p)    ((__attribute__((address_space(3))) async_v4i*)(p))
#endif

// ---------------------------------------------------------------------------
// h[n, j] = embed_table[x[n], j]   (N x 64 gather)
// ---------------------------------------------------------------------------
__global__ void embed_kernel(const int* __restrict__ x,
                             const float* __restrict__ table,
                             float* __restrict__ h, int n_elems) {
    int i = blockIdx.x * blockDim.x + threadIdx.x;
    if (i >= n_elems) return;
    int n = i >> 6;
    int j = i & 63;
    h[i] = table[x[n] * CD + j];
}

__global__ void zero_kernel(float* __restrict__ p, int n) {
    int i = blockIdx.x * blockDim.x + threadIdx.x;
    if (i < n) p[i] = 0.0f;
}

// ---------------------------------------------------------------------------
// Edge phase: one wave32 per edge.
//   theta = (eig[src]-eig[dst]) . w_theta   (butterfly reduce over 8 lanes)
//   lane c rotates complex channel c of h[src], atomic-adds into agg[dst].
// Coalesced 256B gather per wave; 64 f32 L2 atomics per edge (L2-resident).
// ---------------------------------------------------------------------------
__global__ void edge_rotate_scatter(const float* __restrict__ h,
                                    const int* __restrict__ src,
                                    const int* __restrict__ dst,
                                    const float* __restrict__ eigens,
                                    const float* __restrict__ w_theta, // 8 floats
                                    float* __restrict__ agg,
                                    int n_edges) {
    int wave = (blockIdx.x * blockDim.x + threadIdx.x) >> 5;
    int lane = threadIdx.x & 31;
    if (wave >= n_edges) return;

    int s = src[wave];
    int d = dst[wave];

    // theta: each group of 8 lanes computes the full dot product
    int k = lane & 7;
    float diff = eigens[s * ECOMP + k] - eigens[d * ECOMP + k];
    float part = diff * w_theta[k];
    part += __shfl_xor(part, 1);
    part += __shfl_xor(part, 2);
    part += __shfl_xor(part, 4);
    float th = part;

    float c = __cosf(th);
    float sn = __sinf(th);

    const float2 hj = ((const float2*)(h + (size_t)s * CD))[lane];
    float m0 = c * hj.x - sn * hj.y;
    float m1 = sn * hj.x + c * hj.y;

    float* a = agg + (size_t)d * CD + lane * 2;
    atomicAdd(a + 0, m0);
    atomicAdd(a + 1, m1);
}

// ---------------------------------------------------------------------------
// Node GEMM: out[n,o,d] = relu( sum_c agg[n,c,d] * W[c,o] + bias[o] )
// Viewed as (2N x 32) @ (32 x 32) with interleaved A rows (d inner).
// Block = 256 threads = 8 waves; block stages 64 nodes (16KB) + W (4KB) in
// LDS via GLOBAL_LOAD_ASYNC_TO_LDS_B128 (ASYNCcnt path, no VGPR bounce);
// each wave owns 8 nodes (16 A-rows) and produces two 16x16 f32 tiles via
// 16x V_WMMA_F32_16X16X4_F32 (8 K-steps x 2 output col-tiles).
// ---------------------------------------------------------------------------
__global__ void node_gemm_wmma(const float* __restrict__ agg,
                               const float* __restrict__ W,    // 32x32 [c][o]
                               const float* __restrict__ bias, // 32
                               float* __restrict__ hout,
                               int n_nodes) {
    __shared__ float4 sA4[1024];   // 64 nodes * 64 floats = 16KB
    __shared__ float4 sW4[256];    // 32x32 = 4KB

    const int tid = threadIdx.x;
    const float4* agg4 = (const float4*)agg;
    const float4* W4 = (const float4*)W;

#if HAVE_ASYNC_LDS
    // Direct global->LDS async copy (no VGPR round-trip). Out-of-range rows
    // are clamped: WMMA rows are independent and those rows are never stored.
    {
        const long maxfo = (long)n_nodes * 16 - 1;   // last valid float4
#pragma unroll
        for (int i = 0; i < 4; ++i) {
            int off = tid + i * 256;                 // float4 slot in tile
            long fo = (long)blockIdx.x * 1024 + off; // global float4 index
            if (fo > maxfo) fo = maxfo;
            __builtin_amdgcn_global_load_async_to_lds_b128(
                GLOBAL_AS(agg4 + fo), LDS_AS(&sA4[off]), 0, 0);
        }
        __builtin_amdgcn_global_load_async_to_lds_b128(
            GLOBAL_AS(W4 + tid), LDS_AS(&sW4[tid]), 0, 0);
        __builtin_amdgcn_s_wait_asynccnt(0);
    }
#else
    // Fallback: synchronous staging through VGPRs
    sW4[tid] = W4[tid];
#pragma unroll
    for (int i = 0; i < 4; ++i) {
        int off = tid + i * 256;
        long fo = (long)blockIdx.x * 1024 + off;
        int node = (int)(fo >> 4);
        float4 v = make_float4(0.f, 0.f, 0.f, 0.f);
        if (node < n_nodes) v = agg4[fo];
        sA4[off] = v;
    }
#endif
    __syncthreads();

    const float* sW = (const float*)sW4;
    const int wv   = tid >> 5;          // wave in block: 8 nodes each
    const int lane = tid & 31;
    const int m    = lane & 15;         // A row / D col within tile
    const int hi   = lane >> 4;
    const float* awave = ((const float*)sA4) + wv * 512;   // 8 nodes * 64
    const int arow = (m >> 1) * CD + (m & 1);              // row base (d inner)

    v8f acc0 = {0.f, 0.f, 0.f, 0.f, 0.f, 0.f, 0.f, 0.f};
    v8f acc1 = {0.f, 0.f, 0.f, 0.f, 0.f, 0.f, 0.f, 0.f};

#pragma unroll
    for (int st = 0; st < 8; ++st) {
        const int k0 = st * 4 + hi * 2;   // this lane's two K values
        v2f a, b0, b1;
        // A fragment: A[m][k] = agg[node8][k][d] with node8=m>>1, d=m&1
        a.x = awave[arow + (k0 + 0) * 2];
        a.y = awave[arow + (k0 + 1) * 2];
        // B fragment: B[k][n] = W[k*32 + n]
        b0.x = sW[(k0 + 0) * CH + m];
        b0.y = sW[(k0 + 1) * CH + m];
        b1.x = sW[(k0 + 0) * CH + m + 16];
        b1.y = sW[(k0 + 1) * CH + m + 16];
        acc0 = __builtin_amdgcn_wmma_f32_16x16x4_f32(false, a, false, b0,
                                                     (short)0, acc0, false, false);
        acc1 = __builtin_amdgcn_wmma_f32_16x16x4_f32(false, a, false, b1,
                                                     (short)0, acc1, false, false);
    }

    // Bias + ReLU + store (C/D layout: lane col = m, VGPR r -> row hi*8+r)
    const float bias0 = bias[m];
    const float bias1 = bias[m + 16];
    const int nodebase = blockIdx.x * 64 + wv * 8;
#pragma unroll
    for (int r = 0; r < 8; ++r) {
        int rowm = hi * 8 + r;
        int node = nodebase + (rowm >> 1);
        int dd = rowm & 1;
        if (node < n_nodes) {
            float v0 = fmaxf(acc0[r] + bias0, 0.f);
            float v1 = fmaxf(acc1[r] + bias1, 0.f);
            hout[(size_t)node * CD + m * 2 + dd]        = v0;
            hout[(size_t)node * CD + (m + 16) * 2 + dd] = v1;
        }
    }
}

// ---------------------------------------------------------------------------
// Graph pooling: ybuf[batch[n], j] += h[n, j]
// ---------------------------------------------------------------------------
__global__ void pool_kernel(const float* __restrict__ h,
                            const int* __restrict__ batch,
                            float* __restrict__ ybuf, int n_elems) {
    int i = blockIdx.x * blockDim.x + threadIdx.x;
    if (i >= n_elems) return;
    int n = i >> 6;
    int j = i & 63;
    atomicAdd(&ybuf[batch[n] * CD + j], h[i]);
}

// ---------------------------------------------------------------------------
// Final MLP: out[g] = relu(y[g] @ W1 + b1) @ W2 + b2 ; one block, G threads
// ---------------------------------------------------------------------------
__global__ void mlp_kernel(const float* __restrict__ y,
                           const float* __restrict__ W1, // (64,32)
                           const float* __restrict__ b1, // 32
                           const float* __restrict__ W2, // 32
                           const float* __restrict__ b2, // 1
                           float* __restrict__ out) {
    __shared__ float sW1[CD * CH];
    __shared__ float sW2[CH];
    __shared__ float sb1[CH];
    for (int i = threadIdx.x; i < CD * CH; i += blockDim.x) sW1[i] = W1[i];
    if (threadIdx.x < CH) {
        sW2[threadIdx.x] = W2[threadIdx.x];
        sb1[threadIdx.x] = b1[threadIdx.x];
    }
    __syncthreads();

    int g = threadIdx.x;
    float yrow[CD];
#pragma unroll
    for (int i = 0; i < CD; ++i) yrow[i] = y[g * CD + i];

    float acc = b2[0];
#pragma unroll 4
    for (int o = 0; o < CH; ++o) {
        float hsum = sb1[o];
#pragma unroll
        for (int i = 0; i < CD; ++i) hsum = fmaf(yrow[i], sW1[i * CH + o], hsum);
        hsum = fmaxf(hsum, 0.f);
        acc = fmaf(hsum, sW2[o], acc);
    }
    out[g] = acc;
}

// ---------------------------------------------------------------------------
extern "C" void kernel_launch(void* const* d_in, const int* in_sizes, int n_in,
                              void* d_out, int out_size, void* d_ws, size_t ws_size,
                              hipStream_t stream) {
    const int*   x           = (const int*)d_in[0];
    const int*   edge_index  = (const int*)d_in[1];
    const int*   batch       = (const int*)d_in[2];
    const float* eigens      = (const float*)d_in[3];
    const float* embed_table = (const float*)d_in[4];
    const float* w_thetas    = (const float*)d_in[5];   // (L, 8)
    const float* Ws          = (const float*)d_in[6];   // (L, 32, 32)
    const float* bs          = (const float*)d_in[7];   // (L, 32)
    const float* W1          = (const float*)d_in[8];
    const float* b1          = (const float*)d_in[9];
    const float* W2          = (const float*)d_in[10];
    const float* b2          = (const float*)d_in[11];
    float*       out         = (float*)d_out;

    float* h    = (float*)d_ws;                          // N*64
    float* agg  = h + (size_t)N_NODES * CD;              // N*64
    float* ybuf = agg + (size_t)N_NODES * CD;            // G*64

    const int* src = edge_index;
    const int* dst = edge_index + N_EDGES;

    const int nh = N_NODES * CD;                         // 6.4M
    const int grid_nh = (nh + 255) / 256;
    const int grid_edges = (N_EDGES + 7) / 8;            // 1 wave/edge, 8/block
    const int grid_gemm = (N_NODES + 63) / 64;

    // h = embed_table[x]
    embed_kernel<<<grid_nh, 256, 0, stream>>>(x, embed_table, h, nh);

    for (int l = 0; l < NLAYERS; ++l) {
        zero_kernel<<<grid_nh, 256, 0, stream>>>(agg, nh);
        edge_rotate_scatter<<<grid_edges, 256, 0, stream>>>(
            h, src, dst, eigens, w_thetas + l * ECOMP, agg, N_EDGES);
        node_gemm_wmma<<<grid_gemm, 256, 0, stream>>>(
            agg, Ws + l * CH * CH, bs + l * CH, h, N_NODES);
    }

    zero_kernel<<<(NGRAPH * CD + 255) / 256, 256, 0, stream>>>(ybuf, NGRAPH * CD);
    pool_kernel<<<grid_nh, 256, 0, stream>>>(h, batch, ybuf, nh);
    mlp_kernel<<<1, NGRAPH, 0, stream>>>(ybuf, W1, b1, W2, b2, out);
}